// Ops_22170621181973
// MI455X (gfx1250) — compile-verified
//
#include <hip/hip_runtime.h>
#include <hip/hip_bf16.h>

// Problem shape (fixed by setup_inputs in the reference)
constexpr int B_  = 4;
constexpr int N_  = 65536;
constexpr int K_  = 16;
constexpr int D_  = 32;
constexpr int NP_ = N_ / 4;   // 16384

typedef __attribute__((ext_vector_type(4))) unsigned int v4u;
typedef __attribute__((ext_vector_type(8))) int          v8i;
typedef __attribute__((ext_vector_type(4))) int          v4i;

// ---------------------------------------------------------------------------
// TDM: 1-D contiguous tile copy global -> LDS via Tensor Data Mover.
// D# layout per CDNA5 ISA ch.8:
//   group0: [1:0] count=1 | [63:32] lds_addr | [120:64] global_addr | [127:126] type=2
//   group1: [17:16] data_size=2 (4B) | tensor_dim0[79:48] | tensor_dim1[111:80]=1
//           | tile_dim0[127:112] | tile_dim1/2 = 0 (unused) | dim0_stride[207:160]
//   groups 2/3 (+ unused group 4): zero (tile_dim3/4 = 0 => unused)
// This toolchain exposes the 6-arg builtin:
//   (v4u g0, v8i g1, v4i g2, v4i g3, v8i g4, int cpol)
// ---------------------------------------------------------------------------
#if __has_builtin(__builtin_amdgcn_tensor_load_to_lds) && \
    __has_builtin(__builtin_amdgcn_s_wait_tensorcnt)
#define USE_TDM 1
#else
#define USE_TDM 0
#endif

#if USE_TDM
__device__ __forceinline__ void tdm_load_dwords_to_lds(unsigned lds_off,
                                                       unsigned long long ga,
                                                       unsigned ndwords) {
    v4u g0;
    g0[0] = 1u;                                      // count=1, user mode
    g0[1] = lds_off;                                 // lds_addr (bytes)
    g0[2] = (unsigned)ga;                            // global_addr[31:0]
    g0[3] = (unsigned)(ga >> 32) | (2u << 30);       // global_addr[56:32] | type=2
    v8i g1;
    g1[0] = (int)(2u << 16);                         // data_size = 2 -> 4 bytes
    g1[1] = (int)((ndwords & 0xFFFFu) << 16);        // tensor_dim0[15:0]
    g1[2] = (int)(1u << 16);                         // tensor_dim0 hi=0 | tensor_dim1=1
    g1[3] = (int)((ndwords & 0xFFFFu) << 16);        // tensor_dim1 hi=0 | tile_dim0
    g1[4] = 0;                                       // tile_dim1=0, tile_dim2=0 (unused)
    g1[5] = (int)ndwords;                            // tensor_dim0_stride[31:0]
    g1[6] = 0;
    g1[7] = 0;
    v4i z4 = {0, 0, 0, 0};
    v8i z8 = {0, 0, 0, 0, 0, 0, 0, 0};
    __builtin_amdgcn_tensor_load_to_lds(g0, g1, z4, z4, z8, 0);
}
#endif

// ---------------------------------------------------------------------------
// Kernel 1: relative position encoding
//   rel[b,n,k,0:10] = { |c-nb|, (c-nb).xyz, c.xyz, nb.xyz }
// One thread per (b,n,k). The block's 256 contiguous neighbor indices (1 KB)
// are DMA'd into LDS by the TDM (one wave issues, TENSORcnt + barrier guard).
// The 10-float AoS record (40B) is misaligned for wide stores, so results are
// staged in LDS (write stride 10 is bank-conflict-free in wave32) and written
// as 10 fully coalesced non-temporal b32 store waves.
// ---------------------------------------------------------------------------
__global__ __launch_bounds__(256) void relpos_kernel(
    const float* __restrict__ xyz,       // [B,N,3]
    const int*   __restrict__ neigh_idx, // [B,N,K] (flat == thread id)
    float*       __restrict__ rel)       // [B,N,K,10]
{
    __shared__ int   s_idx[256];
    __shared__ float s[256 * 10];

    const int       tid = threadIdx.x;
    const long long t   = (long long)blockIdx.x * 256 + tid;

#if USE_TDM
    if (tid < 32) {   // wave 0 issues one DMA for the whole block's index tile
        const unsigned lds_off = (unsigned)(uintptr_t)(void*)s_idx; // low 32b = LDS offset
        const unsigned long long ga =
            (unsigned long long)(uintptr_t)(neigh_idx + (size_t)blockIdx.x * 256);
        tdm_load_dwords_to_lds(lds_off, ga, 256);
        __builtin_amdgcn_s_wait_tensorcnt(0);
    }
#else
    s_idx[tid] = __builtin_nontemporal_load(neigh_idx + t);
#endif
    __syncthreads();

    const int idx = s_idx[tid];

    const int n = (int)((t >> 4) & (N_ - 1));   // K_ = 16 -> >>4
    const int b = (int)(t >> 20);               // >> (4 + 16)

    const float* cptr = xyz + (size_t)(b * N_ + n) * 3;
    const float* nptr = xyz + ((size_t)b * N_ + (size_t)idx) * 3;

    const float cx = cptr[0], cy = cptr[1], cz = cptr[2];
    const float nx = nptr[0], ny = nptr[1], nz = nptr[2];

    const float rx = cx - nx, ry = cy - ny, rz = cz - nz;
    const float dist = __builtin_amdgcn_sqrtf(rx * rx + ry * ry + rz * rz); // raw v_sqrt_f32

    const float v[10] = {dist, rx, ry, rz, cx, cy, cz, nx, ny, nz};
#pragma unroll
    for (int j = 0; j < 10; ++j)
        s[tid * 10 + j] = v[j];

    __syncthreads();

    float* outb = rel + (size_t)blockIdx.x * (256 * 10);
#pragma unroll
    for (int i = 0; i < 10; ++i)
        __builtin_nontemporal_store(s[i * 256 + tid], outb + i * 256 + tid);
}

// ---------------------------------------------------------------------------
// Kernel 2: random_sample max-pool
//   out[b,d,np] = max_k feat[b,d, pool_idx[b,np,k]]
// One thread per (b,d,np), np fastest -> coalesced output. The 16 indices are
// 64B-aligned -> four b128 loads. Gathers are random 4B but the whole
// feat_pool (33.5 MB) stays resident in the 192 MB L2.
// ---------------------------------------------------------------------------
__global__ __launch_bounds__(256) void maxpool_kernel(
    const float* __restrict__ feat,      // [B,D,N]
    const int*   __restrict__ pool_idx,  // [B,NP,K]
    float*       __restrict__ out)       // [B,D,NP] (flat == thread id)
{
    const long long t    = (long long)blockIdx.x * 256 + threadIdx.x;
    const int       np   = (int)(t & (NP_ - 1));
    const int       rest = (int)(t >> 14);   // log2(NP_) = 14
    const int       di   = rest & (D_ - 1);
    const int       b    = rest >> 5;        // log2(D_) = 5

    const int4* ip = (const int4*)(pool_idx + ((size_t)b * NP_ + np) * K_);
    const int4 i0 = ip[0], i1 = ip[1], i2 = ip[2], i3 = ip[3];

    const float* row = feat + (size_t)(b * D_ + di) * N_;

    float m = row[i0.x];
    m = fmaxf(m, row[i0.y]); m = fmaxf(m, row[i0.z]); m = fmaxf(m, row[i0.w]);
    m = fmaxf(m, row[i1.x]); m = fmaxf(m, row[i1.y]); m = fmaxf(m, row[i1.z]); m = fmaxf(m, row[i1.w]);
    m = fmaxf(m, row[i2.x]); m = fmaxf(m, row[i2.y]); m = fmaxf(m, row[i2.z]); m = fmaxf(m, row[i2.w]);
    m = fmaxf(m, row[i3.x]); m = fmaxf(m, row[i3.y]); m = fmaxf(m, row[i3.z]); m = fmaxf(m, row[i3.w]);

    __builtin_nontemporal_store(m, out + t);
}

// ---------------------------------------------------------------------------
// Kernel 3: nearest interpolation
//   out[b,d,u] = feat[b,d, interp_idx[b,u]]
// One thread per (b,d,u), u fastest -> coalesced output. interp_idx is reused
// 32x across d (L2/L0 hit); feature rows are 64 KB each (L2-resident).
// ---------------------------------------------------------------------------
__global__ __launch_bounds__(256) void interp_kernel(
    const float* __restrict__ feat,        // [B,D,NP]
    const int*   __restrict__ interp_idx,  // [B,N]
    float*       __restrict__ out)         // [B,D,N] (flat == thread id)
{
    const long long t    = (long long)blockIdx.x * 256 + threadIdx.x;
    const int       u    = (int)(t & (N_ - 1));
    const int       rest = (int)(t >> 16);   // log2(N_) = 16
    const int       di   = rest & (D_ - 1);
    const int       b    = rest >> 5;

    const int   idx = interp_idx[(size_t)b * N_ + u];
    const float v   = feat[(size_t)(b * D_ + di) * NP_ + idx];

    __builtin_nontemporal_store(v, out + t);
}

// ---------------------------------------------------------------------------
extern "C" void kernel_launch(void* const* d_in, const int* in_sizes, int n_in,
                              void* d_out, int out_size, void* d_ws, size_t ws_size,
                              hipStream_t stream) {
    (void)in_sizes; (void)n_in; (void)out_size; (void)d_ws; (void)ws_size;

    const float* xyz         = (const float*)d_in[0];  // [B,N,3]
    const int*   neigh_idx   = (const int*)  d_in[1];  // [B,N,K]
    const float* feat_pool   = (const float*)d_in[2];  // [B,D,N,1]
    const int*   pool_idx    = (const int*)  d_in[3];  // [B,NP,K]
    const float* feat_interp = (const float*)d_in[4];  // [B,D,NP,1]
    const int*   interp_idx  = (const int*)  d_in[5];  // [B,N,1]

    float* rel    = (float*)d_out;                                   // B*N*K*10
    float* pooled = rel + (size_t)B_ * N_ * K_ * 10;                 // B*D*NP
    float* interp = pooled + (size_t)B_ * D_ * NP_;                  // B*D*N

    const int n1 = B_ * N_ * K_;   // 4,194,304
    const int n2 = B_ * D_ * NP_;  // 2,097,152
    const int n3 = B_ * D_ * N_;   // 8,388,608

    relpos_kernel <<<n1 / 256, 256, 0, stream>>>(xyz, neigh_idx, rel);
    maxpool_kernel<<<n2 / 256, 256, 0, stream>>>(feat_pool, pool_idx, pooled);
    interp_kernel <<<n3 / 256, 256, 0, stream>>>(feat_interp, interp_idx, interp);
}